// RelativeBucketedTimeAndPositionBasedBias_21440476742331
// MI455X (gfx1250) — compile-verified
//
#include <hip/hip_runtime.h>
#include <stdint.h>

// Problem constants (match reference).
#define B_   16
#define N_   2048
#define NB_  128   // NUM_BUCKETS

typedef float v4f __attribute__((ext_vector_type(4)));
typedef int   v4i __attribute__((ext_vector_type(4)));

// gfx1250 async global->LDS path (CDNA5 async data movement, ASYNCcnt).
#if defined(__gfx1250__) && __has_builtin(__builtin_amdgcn_global_load_async_to_lds_b128)
#define ASYNC_LDS 1
#endif

// ---------------------------------------------------------------------------
// Kernel 1: rel_pos_bias[i, j] = pos_w[(N-1) + j - i]   (1, N, N)
// Pure gather of a 16KB table (stays L1/L2-resident) + streaming b128 NT stores.
// Each thread emits one float4 (4 consecutive j in one row).
// ---------------------------------------------------------------------------
__global__ __launch_bounds__(256) void pos_bias_kernel(const float* __restrict__ pos_w,
                                                       float* __restrict__ out) {
    const int g  = blockIdx.x * 256 + threadIdx.x;   // group of 4 elements
    const int i  = g >> 9;                           // N/4 = 512 groups per row
    const int j0 = (g & 511) << 2;
    const int base = (N_ - 1) + j0 - i;              // in [0, 2N-2]
    v4f v;
    v.x = pos_w[base + 0];
    v.y = pos_w[base + 1];
    v.z = pos_w[base + 2];
    v.w = pos_w[base + 3];
    __builtin_nontemporal_store(v, (v4f*)(out + (size_t)i * N_ + j0));
}

// ---------------------------------------------------------------------------
// Kernel 2: rel_ts_bias[b, i, j] = ts_w[clip(floor(log2(max(|dt|,1))), 0, 128)]
//           dt = ts[b, min(i+1, N-1)] - ts[b, j]
// floor(log2(x)) for integer x >= 1 is exactly 31 - clz(x): no transcendentals.
// Block = 256 threads handles 8 rows of one batch; timestamps row (8KB) and
// ts_w (516B) staged in LDS (async-to-LDS when available).
// ---------------------------------------------------------------------------
__device__ __forceinline__ float bucket_val(int s, int tj, const float* tw) {
    int d = s - tj;
    int a = d < 0 ? -d : d;
    a = a > 1 ? a : 1;              // max(|dt|, 1)
    int bk = 31 - __clz(a);         // floor(log2(a)), exact for ints
    bk = bk > NB_ ? NB_ : bk;       // clip (no-op for |dt| < 2^24, kept for safety)
    return tw[bk];
}

__global__ __launch_bounds__(256) void ts_bias_kernel(const int* __restrict__ ts,
                                                      const float* __restrict__ ts_w,
                                                      float* __restrict__ out) {
    __shared__ __align__(16) int   s_ts[N_];       // 8 KB: one batch's timestamps
    __shared__ float               s_tw[NB_ + 3];  // 129 used

    const int t  = threadIdx.x;
    const int bi = blockIdx.x;       // row tile: rows [bi*8, bi*8+8)
    const int b  = blockIdx.y;

    const int* row = ts + (size_t)b * N_;

#ifdef ASYNC_LDS
    // Two async b128 copies per lane: 256 lanes x 16B x 2 = 8KB row.
    // Builtin signature: (v4i addrspace(1)*, v4i addrspace(3)*, imm offset, imm cpol).
    // uintptr_t round-trip: low 32 bits of a generic LDS pointer are the LDS
    // byte offset (flat->LDS mapping truncates to addr[31:0]), so inttoptr to
    // AS(3) is numerically correct.
    __builtin_amdgcn_global_load_async_to_lds_b128(
        (__attribute__((address_space(1))) v4i*)(uintptr_t)(row + t * 4),
        (__attribute__((address_space(3))) v4i*)(uintptr_t)(&s_ts[t * 4]),
        0, 0);
    __builtin_amdgcn_global_load_async_to_lds_b128(
        (__attribute__((address_space(1))) v4i*)(uintptr_t)(row + 1024 + t * 4),
        (__attribute__((address_space(3))) v4i*)(uintptr_t)(&s_ts[1024 + t * 4]),
        0, 0);
#else
    ((v4i*)s_ts)[t]       = ((const v4i*)row)[t];
    ((v4i*)s_ts)[t + 256] = ((const v4i*)row)[t + 256];
#endif
    if (t <= NB_) s_tw[t] = ts_w[t];
#ifdef ASYNC_LDS
 #if __has_builtin(__builtin_amdgcn_s_wait_asynccnt)
    __builtin_amdgcn_s_wait_asynccnt(0);
 #else
    asm volatile("s_wait_asynccnt 0" ::: "memory");
 #endif
#endif
    __syncthreads();

    // Each thread owns 8 consecutive j, reused across the 8 rows of this tile.
    const int j0 = t * 8;
    const v4i ja = ((const v4i*)(s_ts + j0))[0];
    const v4i jb = ((const v4i*)(s_ts + j0))[1];

    float* outp = out + ((size_t)b * N_ + (size_t)bi * 8) * N_ + j0;

#pragma unroll
    for (int r = 0; r < 8; ++r) {
        const int i  = bi * 8 + r;
        const int ip = (i + 1 < N_) ? (i + 1) : (N_ - 1);
        const int s  = s_ts[ip];     // uniform per iteration -> LDS broadcast

        v4f o0, o1;
        o0.x = bucket_val(s, ja.x, s_tw);
        o0.y = bucket_val(s, ja.y, s_tw);
        o0.z = bucket_val(s, ja.z, s_tw);
        o0.w = bucket_val(s, ja.w, s_tw);
        o1.x = bucket_val(s, jb.x, s_tw);
        o1.y = bucket_val(s, jb.y, s_tw);
        o1.z = bucket_val(s, jb.z, s_tw);
        o1.w = bucket_val(s, jb.w, s_tw);

        __builtin_nontemporal_store(o0, (v4f*)(outp));
        __builtin_nontemporal_store(o1, (v4f*)(outp + 4));
        outp += N_;
    }
}

// ---------------------------------------------------------------------------
// Inputs (setup_inputs order): [0] all_timestamps int32 (B,N)
//                              [1] pos_w  f32 (2N-1,)
//                              [2] ts_w   f32 (NUM_BUCKETS+1,)
// Output: rel_pos_bias (1,N,N) followed by rel_ts_bias (B,N,N), f32 flat.
// ---------------------------------------------------------------------------
extern "C" void kernel_launch(void* const* d_in, const int* in_sizes, int n_in,
                              void* d_out, int out_size, void* d_ws, size_t ws_size,
                              hipStream_t stream) {
    const int*   ts    = (const int*)d_in[0];
    const float* pos_w = (const float*)d_in[1];
    const float* ts_w  = (const float*)d_in[2];
    float*       out   = (float*)d_out;

    // rel_pos_bias: N*N/4 float4 groups, 256 threads/block -> 4096 blocks.
    pos_bias_kernel<<<dim3((N_ * N_ / 4) / 256), dim3(256), 0, stream>>>(pos_w, out);

    // rel_ts_bias: grid (N/8 row-tiles, B batches), 256 threads (8 wave32).
    ts_bias_kernel<<<dim3(N_ / 8, B_), dim3(256), 0, stream>>>(ts, ts_w, out + (size_t)N_ * N_);

    (void)in_sizes; (void)n_in; (void)out_size; (void)d_ws; (void)ws_size;
}